// CGWeight_4870492914235
// MI455X (gfx1250) — compile-verified
//
#include <hip/hip_runtime.h>

// ======================================================================
// Compile-time Clebsch-Gordan machinery (mirrors the Python reference).
// Everything is evaluated by clang at compile time into a __constant__
// table; the kernel itself is a single wave32 doing 136 chained
// v_wmma_f32_16x16x4_f32 ops (full fp32 precision).
// ======================================================================
namespace cgc {

constexpr double fact(int n) { double r = 1.0; for (int i = 2; i <= n; ++i) r *= (double)i; return r; }

constexpr double csqrt(double x) {
  if (x <= 0.0) return 0.0;
  double g = (x >= 1.0) ? x : 1.0;
  for (int i = 0; i < 64; ++i) g = 0.5 * (g + x / g);
  return g;
}

constexpr double neg1pow(int n) { return (n & 1) ? -1.0 : 1.0; }

// Racah formula, verbatim from the reference.
constexpr double su2_cg(int j1, int m1, int j2, int m2, int j3, int m3) {
  if (m3 != m1 + m2) return 0.0;
  int vmin = 0;
  if (-j1 + j2 + m3 > vmin) vmin = -j1 + j2 + m3;
  if (-j1 + m1      > vmin) vmin = -j1 + m1;
  int vmax = j2 + j3 + m1;
  if (j3 - j1 + j2 < vmax) vmax = j3 - j1 + j2;
  if (j3 + m3      < vmax) vmax = j3 + m3;
  double C = csqrt((2.0 * j3 + 1.0) * fact(j3 + j1 - j2) * fact(j3 - j1 + j2) * fact(j1 + j2 - j3)
                   / fact(j1 + j2 + j3 + 1)
                   * fact(j3 + m3) * fact(j3 - m3)
                   / (fact(j1 + m1) * fact(j1 - m1) * fact(j2 + m2) * fact(j2 - m2)));
  double S = 0.0;
  for (int v = vmin; v <= vmax; ++v) {
    S += neg1pow(v + j2 + m2) * fact(j2 + j3 + m1 - v) * fact(j1 - m1 + v)
         / (fact(v) * fact(j3 - j1 + j2 - v) * fact(j3 + m3 - v) * fact(v + j1 - j2 - m3));
  }
  return C * S;
}

struct cx { double re, im; };
constexpr cx cmul(cx a, cx b) { return {a.re * b.re - a.im * b.im, a.re * b.im + a.im * b.re}; }
constexpr cx cadd(cx a, cx b) { return {a.re + b.re, a.im + b.im}; }
constexpr cx cscale(cx a, double s) { return {a.re * s, a.im * s}; }
constexpr cx cconj(cx a) { return {a.re, -a.im}; }

struct QMat {
  cx  q[9][9];
  int nnz[9];
  int col[9][2];
  cx  val[9][2];
};

// e3nn real -> complex change of basis, times (-i)^l.
constexpr QMat change_basis(int l) {
  QMat Q{};
  const double is2 = 1.0 / csqrt(2.0);
  for (int m = -l; m <= l; ++m) {
    if (m < 0) {
      Q.q[l + m][l - m] = cx{is2, 0.0};     // col l+|m|
      Q.q[l + m][l + m] = cx{0.0, -is2};    // col l-|m|
    } else if (m == 0) {
      Q.q[l][l] = cx{1.0, 0.0};
    } else {
      const double s = neg1pow(m) * is2;
      Q.q[l + m][l + m] = cx{s, 0.0};
      Q.q[l + m][l - m] = cx{0.0, s};
    }
  }
  const int r4 = l % 4;
  const cx ph = (r4 == 0) ? cx{1.0, 0.0} : (r4 == 1) ? cx{0.0, -1.0}
              : (r4 == 2) ? cx{-1.0, 0.0} : cx{0.0, 1.0};
  const int n = 2 * l + 1;
  for (int i = 0; i < n; ++i)
    for (int j = 0; j < n; ++j)
      Q.q[i][j] = cmul(ph, Q.q[i][j]);
  for (int i = 0; i < n; ++i) {
    Q.nnz[i] = 0;
    for (int j = 0; j < n; ++j)
      if (Q.q[i][j].re != 0.0 || Q.q[i][j].im != 0.0) {
        Q.col[i][Q.nnz[i]] = j;
        Q.val[i][Q.nnz[i]] = Q.q[i][j];
        Q.nnz[i] += 1;
      }
  }
  return Q;
}

struct CGT { double c[9][9][5]; };

// Real SO(3) CG: einsum('ij,kl,mn,ikn->jlm', Q1, Q2, conj(Q3.T), Csu2).real
// factored into three sparse contractions to stay within constexpr limits.
constexpr CGT so3_cg(int l1, int l2) {
  const int l3 = 2;
  const int n1 = 2 * l1 + 1, n2 = 2 * l2 + 1, n3 = 2 * l3 + 1;
  const QMat Q1 = change_basis(l1);
  const QMat Q2 = change_basis(l2);
  const QMat Q3 = change_basis(l3);

  cx T1[9][9][5]{};                       // T1[j][k][n] = sum_i Q1[i][j] * Csu2[i][k][n]
  for (int i = 0; i < n1; ++i) {
    const int m1 = i - l1;
    for (int k = 0; k < n2; ++k) {
      const int m2 = k - l2;
      const int m3 = m1 + m2;
      if (m3 < -l3 || m3 > l3) continue;
      const double c = su2_cg(l1, m1, l2, m2, l3, m3);
      if (c == 0.0) continue;
      const int nn = l3 + m3;
      for (int t = 0; t < Q1.nnz[i]; ++t) {
        const int j = Q1.col[i][t];
        T1[j][k][nn] = cadd(T1[j][k][nn], cscale(Q1.val[i][t], c));
      }
    }
  }

  cx T2[9][9][5]{};                       // T2[j][l][n] = sum_k Q2[k][l] * T1[j][k][n]
  for (int k = 0; k < n2; ++k)
    for (int t = 0; t < Q2.nnz[k]; ++t) {
      const int lc = Q2.col[k][t];
      const cx  qv = Q2.val[k][t];
      for (int j = 0; j < n1; ++j)
        for (int n = 0; n < n3; ++n) {
          const cx v = T1[j][k][n];
          if (v.re != 0.0 || v.im != 0.0)
            T2[j][lc][n] = cadd(T2[j][lc][n], cmul(qv, v));
        }
    }

  CGT R{};                                // R[j][l][m] = Re( sum_n conj(Q3[n][m]) * T2[j][l][n] )
  for (int n = 0; n < n3; ++n)
    for (int t = 0; t < Q3.nnz[n]; ++t) {
      const int m  = Q3.col[n][t];
      const cx  qv = cconj(Q3.val[n][t]);
      for (int j = 0; j < n1; ++j)
        for (int lc = 0; lc < n2; ++lc) {
          const cx v = T2[j][lc][n];
          if (v.re != 0.0 || v.im != 0.0)
            R.c[j][lc][m] += qv.re * v.re - qv.im * v.im;
        }
    }
  return R;
}

// ---- packed device tables -------------------------------------------------
// x[n] = weight[c] * A25[i] * H25[j] for 532 valid triples (pad to 544).
// out[k] = sum_n G[k][n] * x[n]  -> chained V_WMMA_F32_16X16X4_F32:
//   chunk c (K=4): A lane L = Gpad[L&15][4c + 2*(L>>4) + {0,1}]  (exact ISA layout)
//                  B lane L = x[4c + 2*(L>>4) + {0,1}] (all 16 N columns replicated)
struct Tables {
  alignas(16) float apack[136][32][2];  // 136 chunks * 32 lanes * 2 VGPRs
  unsigned short idx[544];              // (combo<<10)|(aIdx<<5)|hIdx ; 0xFFFF = pad
  unsigned char  seg[32];               // ragged-input gather: which a_l / h_l
  unsigned char  off[32];               // offset inside that segment
};

constexpr Tables make_tables() {
  Tables t{};
  {
    int p = 0;
    for (int l = 0; l < 5; ++l)
      for (int m = 0; m < 2 * l + 1; ++m) {
        t.seg[p] = (unsigned char)l;
        t.off[p] = (unsigned char)m;
        ++p;
      }
  }
  double G[5][544]{};
  int n = 0, cidx = 0;
  for (int la = 0; la < 5; ++la)
    for (int lh = 0; lh < 5; ++lh) {
      int d = la - lh; if (d < 0) d = -d;
      if (!(d <= 2 && 2 <= la + lh)) continue;           // triangle inequality, L_OUT=2
      const CGT cg = so3_cg(la, lh);
      const int ab = la * la, hb = lh * lh;              // prefix offsets: sum(2l'+1) = l^2
      for (int i = 0; i < 2 * la + 1; ++i)
        for (int j = 0; j < 2 * lh + 1; ++j) {
          t.idx[n] = (unsigned short)((cidx << 10) | ((ab + i) << 5) | (hb + j));
          for (int k = 0; k < 5; ++k) G[k][n] = cg.c[i][j][k];
          ++n;
        }
      ++cidx;
    }
  for (int m = n; m < 544; ++m) t.idx[m] = 0xFFFFu;
  for (int c = 0; c < 136; ++c)
    for (int L = 0; L < 32; ++L) {
      const int row = L & 15;
      const int kb  = 4 * c + 2 * (L >> 4);
      t.apack[c][L][0] = (row < 5) ? (float)G[row][kb + 0] : 0.0f;
      t.apack[c][L][1] = (row < 5) ? (float)G[row][kb + 1] : 0.0f;
    }
  return t;
}

} // namespace cgc

__device__ __constant__ cgc::Tables gT = cgc::make_tables();

typedef float v2f __attribute__((ext_vector_type(2)));
typedef float v8f __attribute__((ext_vector_type(8)));

__global__ __launch_bounds__(32) void CGWeight_kernel(
    const float* __restrict__ a0, const float* __restrict__ a1, const float* __restrict__ a2,
    const float* __restrict__ a3, const float* __restrict__ a4,
    const float* __restrict__ h0, const float* __restrict__ h1, const float* __restrict__ h2,
    const float* __restrict__ h3, const float* __restrict__ h4,
    const float* __restrict__ w,  float* __restrict__ out) {
  __shared__ float A25[32];
  __shared__ float H25[32];
  __shared__ float W16[16];
  __shared__ __align__(16) float X[544];

  const int lane = threadIdx.x;

  // Gather the 11 ragged inputs into LDS (single wave -> cheap).
  if (lane < 25) {
    const int s = gT.seg[lane];
    const int o = gT.off[lane];
    const float* ap = (s == 0) ? a0 : (s == 1) ? a1 : (s == 2) ? a2 : (s == 3) ? a3 : a4;
    const float* hp = (s == 0) ? h0 : (s == 1) ? h1 : (s == 2) ? h2 : (s == 3) ? h3 : h4;
    A25[lane] = ap[o];
    H25[lane] = hp[o];
  }
  if (lane < 16) W16[lane] = w[lane];
  __syncthreads();

  // x[n] = w_c * a_i * h_j  (544 = 17 * 32, zero-padded)
#pragma unroll
  for (int r = 0; r < 17; ++r) {
    const int n = r * 32 + lane;
    const unsigned e = gT.idx[n];
    float v = 0.0f;
    if (e != 0xFFFFu)
      v = W16[e >> 10] * A25[(e >> 5) & 31] * H25[e & 31];
    X[n] = v;
  }
  __syncthreads();

  // out[k] = sum_n G[k][n] x[n] via 136 chained fp32 WMMAs, 4 accumulator
  // chains to pipeline XDL latency. All 16 B columns carry replicated x,
  // only column N=0 of D is consumed. EXEC is all-ones here (one full wave).
  v8f acc0 = {}, acc1 = {}, acc2 = {}, acc3 = {};
  const int xo = (lane >> 4) * 2;  // 0 for lanes 0-15 (K=0,1), 2 for lanes 16-31 (K=2,3)
  for (int c = 0; c < 136; c += 4) {
    const v2f aA = *(const v2f*)(&gT.apack[c + 0][lane][0]);
    const v2f bA = *(const v2f*)(&X[4 * (c + 0) + xo]);
    acc0 = __builtin_amdgcn_wmma_f32_16x16x4_f32(false, aA, false, bA, (short)0, acc0, false, false);

    const v2f aB = *(const v2f*)(&gT.apack[c + 1][lane][0]);
    const v2f bB = *(const v2f*)(&X[4 * (c + 1) + xo]);
    acc1 = __builtin_amdgcn_wmma_f32_16x16x4_f32(false, aB, false, bB, (short)0, acc1, false, false);

    const v2f aC = *(const v2f*)(&gT.apack[c + 2][lane][0]);
    const v2f bC = *(const v2f*)(&X[4 * (c + 2) + xo]);
    acc2 = __builtin_amdgcn_wmma_f32_16x16x4_f32(false, aC, false, bC, (short)0, acc2, false, false);

    const v2f aD = *(const v2f*)(&gT.apack[c + 3][lane][0]);
    const v2f bD = *(const v2f*)(&X[4 * (c + 3) + xo]);
    acc3 = __builtin_amdgcn_wmma_f32_16x16x4_f32(false, aD, false, bD, (short)0, acc3, false, false);
  }
  const v8f acc = (acc0 + acc1) + (acc2 + acc3);

  // D layout: lane 0, VGPR k -> D[M=k][N=0] = out[k] for k=0..4.
  if (lane == 0) {
#pragma unroll
    for (int k = 0; k < 5; ++k) {
      float v = acc[k];
      v = (v != v) ? 0.0f : v;                                   // nan -> 0
      if (v >  3.4028234663852886e38f) v =  3.4028234663852886e38f;  // +inf clamp
      if (v < -3.4028234663852886e38f) v = -3.4028234663852886e38f;  // -inf clamp
      out[k] = v;
    }
  }
}

extern "C" void kernel_launch(void* const* d_in, const int* in_sizes, int n_in,
                              void* d_out, int out_size, void* d_ws, size_t ws_size,
                              hipStream_t stream) {
  (void)in_sizes; (void)n_in; (void)out_size; (void)d_ws; (void)ws_size;
  const float* a0 = (const float*)d_in[0];
  const float* a1 = (const float*)d_in[1];
  const float* a2 = (const float*)d_in[2];
  const float* a3 = (const float*)d_in[3];
  const float* a4 = (const float*)d_in[4];
  const float* h0 = (const float*)d_in[5];
  const float* h1 = (const float*)d_in[6];
  const float* h2 = (const float*)d_in[7];
  const float* h3 = (const float*)d_in[8];
  const float* h4 = (const float*)d_in[9];
  const float* w  = (const float*)d_in[10];
  CGWeight_kernel<<<dim3(1), dim3(32), 0, stream>>>(
      a0, a1, a2, a3, a4, h0, h1, h2, h3, h4, w, (float*)d_out);
}